// GIN2_57071525429596
// MI455X (gfx1250) — compile-verified
//
#include <hip/hip_runtime.h>
#include <hip/hip_bf16.h>

#define NN 100000
#define NE 1600000
#define DD 128
#define NL 3
#define NC 10
#define NG 512
#define BN_EPS 1e-5f

// Scatter kernel geometry: fixed wave count, grid-stride over edges.
#define SC_BLOCKS 25000
#define SC_WAVES (SC_BLOCKS * 8)

typedef __attribute__((ext_vector_type(2))) float v2f;
typedef __attribute__((ext_vector_type(8))) float v8f;

// ---------------------------------------------------------------------------
// Edge aggregation: agg[dst] += x[src].  One wave per edge-slot, lane covers
// 4 floats; grid-stride loop with per-lane global_prefetch of the NEXT edge's
// source row (x is L2-resident: 51 MB << 192 MB L2), so gathers are
// prefetch-covered L2 hits while the current edge's f32 atomics drain.
// ---------------------------------------------------------------------------
__global__ __launch_bounds__(256) void scatter_add_kernel(
    const float* __restrict__ x, const int* __restrict__ src,
    const int* __restrict__ dst, float* __restrict__ agg) {
  const int lane = threadIdx.x & 31;
  int e = blockIdx.x * 8 + (threadIdx.x >> 5);
  for (; e < NE; e += SC_WAVES) {
    const int nexte = e + SC_WAVES;
    if (nexte < NE) {
      const int ns = src[nexte];
      __builtin_prefetch(x + (size_t)ns * DD + lane * 4, 0, 0);
    }
    const int s = src[e];
    const int d = dst[e];
    const float4 v = *(const float4*)(x + (size_t)s * DD + lane * 4);
    float* o = agg + (size_t)d * DD + lane * 4;
    atomicAdd(o + 0, v.x);
    atomicAdd(o + 1, v.y);
    atomicAdd(o + 2, v.z);
    atomicAdd(o + 3, v.w);
  }
}

// ---------------------------------------------------------------------------
// out = relu((inA [+ inB]) @ W + bias), M=NN rows, K=N=128, exact fp32 via
// V_WMMA_F32_16X16X4_F32.  Block = 512 threads = 16 waves -> 32x128 out tile
// (2 row sub-tiles x 8 col tiles).  A tile + 64-row W chunks staged in LDS.
// Two independent accumulator chains (even/odd K-steps) halve the WMMA RAW
// dependency depth; summed exactly in the epilogue.
// Safe to run in-place (out==inA): all reads staged to LDS before any write.
// ---------------------------------------------------------------------------
__global__ __launch_bounds__(512) void gin_gemm_kernel(
    const float* __restrict__ inA, const float* __restrict__ inB,
    const float* __restrict__ W, const float* __restrict__ bias,
    float* __restrict__ out) {
  __shared__ float As[32][DD + 1];
  __shared__ float Ws[64][DD + 1];

  const int tid = threadIdx.x;
  const size_t row0 = (size_t)blockIdx.x * 32;

  // Stage A tile (optionally fused add of second operand): 4096 elems / 512 thr
  for (int i = tid; i < 32 * DD; i += 512) {
    const size_t gi = row0 * DD + (size_t)i;
    float v = inA[gi];
    if (inB) v += inB[gi];
    As[i >> 7][i & 127] = v;
  }

  const int lane = tid & 31;
  const int wid  = tid >> 5;     // 0..15
  const int rt   = wid >> 3;     // row sub-tile 0..1
  const int ct   = wid & 7;      // col tile 0..7
  const int half = lane >> 4;    // K-half select per ISA layout
  const int mrow = rt * 16 + (lane & 15);
  const int ncol = ct * 16 + (lane & 15);

  v8f c0 = {};
  v8f c1 = {};
  for (int kc = 0; kc < DD; kc += 64) {
    // Stage 64 rows of W: 8192 elems / 512 thr
    __syncthreads();
    for (int i = tid; i < 64 * DD; i += 512) {
      Ws[i >> 7][i & 127] = W[(size_t)kc * DD + (size_t)i];
    }
    __syncthreads();

    for (int kb = 0; kb < 64; kb += 8) {
      // A 16x4 f32 fragment: VGPR v holds K = kb + half*2 + v   (ISA 7.12.2)
      v2f a0, b0, a1, b1;
      a0[0] = As[mrow][kc + kb + half * 2 + 0];
      a0[1] = As[mrow][kc + kb + half * 2 + 1];
      // B 4x16 f32 fragment: lane -> N column, same K split
      b0[0] = Ws[kb + half * 2 + 0][ncol];
      b0[1] = Ws[kb + half * 2 + 1][ncol];
      a1[0] = As[mrow][kc + kb + 4 + half * 2 + 0];
      a1[1] = As[mrow][kc + kb + 4 + half * 2 + 1];
      b1[0] = Ws[kb + 4 + half * 2 + 0][ncol];
      b1[1] = Ws[kb + 4 + half * 2 + 1][ncol];
      c0 = __builtin_amdgcn_wmma_f32_16x16x4_f32(
          false, a0, false, b0, (short)0, c0, false, false);
      c1 = __builtin_amdgcn_wmma_f32_16x16x4_f32(
          false, a1, false, b1, (short)0, c1, false, false);
    }
  }

  // Epilogue: merge chains, bias + ReLU.
  // C layout: VGPR v -> M = v + half*8, lane&15 -> N.
  const float bv = bias[ncol];
  const size_t rbase = row0 + (size_t)rt * 16 + (size_t)half * 8;
#pragma unroll
  for (int v = 0; v < 8; ++v) {
    float r = c0[v] + c1[v] + bv;
    r = r > 0.0f ? r : 0.0f;
    out[(rbase + v) * DD + ncol] = r;
  }
}

// ---------------------------------------------------------------------------
// Column sums / sums-of-squares for BatchNorm (training-mode batch stats).
// blockDim = 128 (one thread per column), each block covers a row stripe.
// ---------------------------------------------------------------------------
__global__ __launch_bounds__(128) void col_stats_kernel(
    const float* __restrict__ h, float* __restrict__ stats) {
  const int c = threadIdx.x;
  int r0 = blockIdx.x * 256;
  int r1 = r0 + 256;
  if (r1 > NN) r1 = NN;
  float s = 0.0f, s2 = 0.0f;
  for (int r = r0; r < r1; ++r) {
    const float v = h[(size_t)r * DD + c];
    s += v;
    s2 += v * v;
  }
  atomicAdd(&stats[c], s);
  atomicAdd(&stats[DD + c], s2);
}

__global__ __launch_bounds__(256) void bn_apply_kernel(
    const float* __restrict__ h, const float* __restrict__ stats,
    const float* __restrict__ gamma, const float* __restrict__ beta,
    float* __restrict__ out) {
  const int idx = blockIdx.x * 256 + threadIdx.x;
  if (idx >= NN * DD) return;
  const int c = idx & 127;
  const float inv_n = 1.0f / (float)NN;
  const float mu = stats[c] * inv_n;
  const float var = stats[DD + c] * inv_n - mu * mu;
  const float sc = gamma[c] * rsqrtf(var + BN_EPS);
  out[idx] = (h[idx] - mu) * sc + beta[c];
}

// ---------------------------------------------------------------------------
// Global mean pool: atomic segment-sum into pooled[g] + per-graph counts.
// ---------------------------------------------------------------------------
__global__ __launch_bounds__(256) void pool_kernel(
    const float* __restrict__ x, const int* __restrict__ batch,
    float* __restrict__ pooled, float* __restrict__ counts) {
  const int idx = blockIdx.x * 256 + threadIdx.x;
  if (idx >= NN * DD) return;
  const int r = idx >> 7;
  const int c = idx & 127;
  const int g = batch[r];
  atomicAdd(&pooled[(size_t)g * DD + c], x[idx]);
  if (c == 0) atomicAdd(&counts[g], 1.0f);
}

// ---------------------------------------------------------------------------
// Classifier head: one block per graph.  h = relu(pooled@lin1_w + b1);
// out = h@lin2_w + b2.  512x(128x128 + 128x10) dots: negligible work.
// ---------------------------------------------------------------------------
__global__ __launch_bounds__(128) void head_kernel(
    const float* __restrict__ pooled, const float* __restrict__ counts,
    const float* __restrict__ lin1_w, const float* __restrict__ lin1_b,
    const float* __restrict__ lin2_w, const float* __restrict__ lin2_b,
    float* __restrict__ out) {
  __shared__ float p[DD];
  __shared__ float h1[DD];
  const int g = blockIdx.x;
  const int t = threadIdx.x;
  float cnt = counts[g];
  cnt = cnt > 1.0f ? cnt : 1.0f;
  p[t] = pooled[(size_t)g * DD + t] / cnt;
  __syncthreads();
  float acc = lin1_b[t];
#pragma unroll 8
  for (int k = 0; k < DD; ++k) acc += p[k] * lin1_w[(size_t)k * DD + t];
  h1[t] = acc > 0.0f ? acc : 0.0f;
  __syncthreads();
  if (t < NC) {
    float o = lin2_b[t];
#pragma unroll 8
    for (int k = 0; k < DD; ++k) o += h1[k] * lin2_w[(size_t)k * NC + t];
    out[(size_t)g * NC + t] = o;
  }
}

// ---------------------------------------------------------------------------
extern "C" void kernel_launch(void* const* d_in, const int* in_sizes, int n_in,
                              void* d_out, int out_size, void* d_ws, size_t ws_size,
                              hipStream_t stream) {
  const float* x       = (const float*)d_in[0];
  const int*   ei      = (const int*)d_in[1];
  const int*   batch   = (const int*)d_in[2];
  const float* W1s     = (const float*)d_in[3];
  const float* b1s     = (const float*)d_in[4];
  const float* W2s     = (const float*)d_in[5];
  const float* b2s     = (const float*)d_in[6];
  const float* gammas  = (const float*)d_in[7];
  const float* betas   = (const float*)d_in[8];
  const float* lin1_w  = (const float*)d_in[9];
  const float* lin1_b  = (const float*)d_in[10];
  const float* lin2_w  = (const float*)d_in[11];
  const float* lin2_b  = (const float*)d_in[12];
  float* outp = (float*)d_out;

  // Workspace partition (only 2 large buffers: agg + xbuf = ~102 MB)
  float* agg    = (float*)d_ws;                  // NN*DD
  float* xbuf   = agg + (size_t)NN * DD;         // NN*DD
  float* stats  = xbuf + (size_t)NN * DD;        // 2*DD
  float* pooled = stats + 2 * DD;                // NG*DD
  float* counts = pooled + (size_t)NG * DD;      // NG

  const int* src = ei;        // edge_index[0]
  const int* dst = ei + NE;   // edge_index[1]

  const float* xcur = x;
  for (int l = 0; l < NL; ++l) {
    hipMemsetAsync(agg, 0, (size_t)NN * DD * sizeof(float), stream);
    scatter_add_kernel<<<SC_BLOCKS, 256, 0, stream>>>(xcur, src, dst, agg);
    // h1 = relu((x + agg) @ W1 + b1)   (in-place into agg; reads staged first)
    gin_gemm_kernel<<<NN / 32, 512, 0, stream>>>(
        agg, xcur, W1s + (size_t)l * DD * DD, b1s + (size_t)l * DD, agg);
    // h2 = relu(h1 @ W2 + b2) -> xbuf  (GEMM2 never reads xcur, so safe
    //  even when xcur == xbuf on layers >= 1)
    gin_gemm_kernel<<<NN / 32, 512, 0, stream>>>(
        agg, nullptr, W2s + (size_t)l * DD * DD, b2s + (size_t)l * DD, xbuf);
    // BatchNorm (batch stats, biased var), in-place on xbuf
    hipMemsetAsync(stats, 0, 2 * DD * sizeof(float), stream);
    col_stats_kernel<<<(NN + 255) / 256, 128, 0, stream>>>(xbuf, stats);
    bn_apply_kernel<<<(NN * DD) / 256, 256, 0, stream>>>(
        xbuf, stats, gammas + (size_t)l * DD, betas + (size_t)l * DD, xbuf);
    xcur = xbuf;
  }

  hipMemsetAsync(pooled, 0, ((size_t)NG * DD + NG) * sizeof(float), stream);
  pool_kernel<<<(NN * DD) / 256, 256, 0, stream>>>(xcur, batch, pooled, counts);
  head_kernel<<<NG, DD, 0, stream>>>(pooled, counts, lin1_w, lin1_b,
                                     lin2_w, lin2_b, outp);
}